// HNHN_31842887533233
// MI455X (gfx1250) — compile-verified
//
#include <hip/hip_runtime.h>

#define N_NODES 100000
#define M_EDGES 40000
#define E_INC   640000
#define DIM     128

typedef float v2f __attribute__((ext_vector_type(2)));
typedef float v8f __attribute__((ext_vector_type(8)));

#define ROWS_PER_BLOCK 64
#define ASTRIDE 132   // (132*m) % 64 = 4m -> conflict-free column reads of A tile
#define WSTRIDE 144   // lane-group row offset 2*144 % 64 = 32 -> disjoint banks

// out[r, c] = rowscale[r] * ( (RELU? max(A,0) : A)[r,:] @ W[:,c] + bias[c] )
template <bool RELU>
__global__ __launch_bounds__(256) void hnhn_gemm_kernel(
    const float* __restrict__ A, const float* __restrict__ W,
    const float* __restrict__ bias, const float* __restrict__ rowscale,
    float* __restrict__ out, int n_rows)
{
    __shared__ float As[ROWS_PER_BLOCK * ASTRIDE];
    __shared__ float Ws[DIM * WSTRIDE];

    const int tid = threadIdx.x;
    const int rowBase = blockIdx.x * ROWS_PER_BLOCK;

    // Stage W (128x128 fp32) into LDS, padded stride.
    #pragma unroll
    for (int i = 0; i < 16; ++i) {
        int linear = i * 256 + tid;          // 0..4095 over 128 rows x 32 qcols
        int row = linear >> 5;
        int qc  = (linear & 31) << 2;
        float4 v = *(const float4*)(W + row * DIM + qc);
        *(float4*)(&Ws[row * WSTRIDE + qc]) = v;
    }
    // Stage A tile (<=64 rows x 128), optional ReLU, zero-padded OOB rows.
    #pragma unroll
    for (int i = 0; i < 8; ++i) {
        int linear = i * 256 + tid;          // 0..2047 over 64 rows x 32 qcols
        int row = linear >> 5;
        int qc  = (linear & 31) << 2;
        float4 v = make_float4(0.f, 0.f, 0.f, 0.f);
        int gr = rowBase + row;
        if (gr < n_rows) v = *(const float4*)(A + (size_t)gr * DIM + qc);
        if (RELU) {
            v.x = fmaxf(v.x, 0.f); v.y = fmaxf(v.y, 0.f);
            v.z = fmaxf(v.z, 0.f); v.w = fmaxf(v.w, 0.f);
        }
        *(float4*)(&As[row * ASTRIDE + qc]) = v;
    }
    __syncthreads();

    const int wave = tid >> 5;        // 0..7 -> column tile
    const int lane = tid & 31;
    const int lg   = lane >> 4;       // lane group (0: lanes 0-15, 1: lanes 16-31)
    const int ln   = lane & 15;
    const int colBase = wave * 16;

    v8f acc[4];
    acc[0] = v8f{}; acc[1] = v8f{}; acc[2] = v8f{}; acc[3] = v8f{};

    // K loop: V_WMMA_F32_16X16X4_F32, 32 steps of K=4.
    #pragma unroll 4
    for (int k0 = 0; k0 < DIM; k0 += 4) {
        const int kb = k0 + 2 * lg;   // A/B frag: K = vgpr + 2*lanegroup
        v2f b;
        b[0] = Ws[(kb + 0) * WSTRIDE + colBase + ln];
        b[1] = Ws[(kb + 1) * WSTRIDE + colBase + ln];
        #pragma unroll
        for (int r = 0; r < 4; ++r) {
            const float* ap = &As[(r * 16 + ln) * ASTRIDE + kb];
            v2f a;
            a[0] = ap[0];
            a[1] = ap[1];
            acc[r] = __builtin_amdgcn_wmma_f32_16x16x4_f32(
                false, a, false, b, (short)0, acc[r], false, false);
        }
    }

    // Epilogue: bias add then row scale; C/D layout M = vgpr + 8*lanegroup.
    const float bcol = bias[colBase + ln];
    if (rowBase + ROWS_PER_BLOCK <= n_rows) {
        // Fast path: whole tile in range, no per-element guards.
        #pragma unroll
        for (int r = 0; r < 4; ++r) {
            #pragma unroll
            for (int v = 0; v < 8; ++v) {
                int grow = rowBase + r * 16 + v + 8 * lg;
                out[(size_t)grow * DIM + colBase + ln] =
                    rowscale[grow] * (acc[r][v] + bcol);
            }
        }
    } else {
        #pragma unroll
        for (int r = 0; r < 4; ++r) {
            #pragma unroll
            for (int v = 0; v < 8; ++v) {
                int grow = rowBase + r * 16 + v + 8 * lg;
                if (grow < n_rows) {
                    out[(size_t)grow * DIM + colBase + ln] =
                        rowscale[grow] * (acc[r][v] + bcol);
                }
            }
        }
    }
}

// dst[sidx[e], :] += scale[sidx[e]] * src[gidx[e], :]
// One wave per incidence entry; each lane owns a float4 of the 128-wide row.
// Indices are wave-uniform: force them into SGPRs so the gather becomes
// saddr-based global_load_b128 and the scale fetch becomes a scalar load.
__global__ __launch_bounds__(256) void hnhn_scatter_kernel(
    const float* __restrict__ src, const int* __restrict__ gidx,
    const int* __restrict__ sidx, const float* __restrict__ scale,
    float* __restrict__ dst, int nE)
{
    int entry = blockIdx.x * 8 + (threadIdx.x >> 5);
    int lane  = threadIdx.x & 31;
    if (entry >= nE) return;
    int g = __builtin_amdgcn_readfirstlane(gidx[entry]);
    int s = __builtin_amdgcn_readfirstlane(sidx[entry]);
    float sc = scale[s];
    const float4* srow = (const float4*)(src + (size_t)g * DIM);
    float4 v = srow[lane];
    float* d = dst + (size_t)s * DIM + lane * 4;
    unsafeAtomicAdd(d + 0, sc * v.x);
    unsafeAtomicAdd(d + 1, sc * v.y);
    unsafeAtomicAdd(d + 2, sc * v.z);
    unsafeAtomicAdd(d + 3, sc * v.w);
}

__global__ __launch_bounds__(256) void hnhn_zero_kernel(float* __restrict__ p, int n)
{
    int i = (blockIdx.x * 256 + threadIdx.x) * 4;
    if (i < n) *(float4*)(p + i) = make_float4(0.f, 0.f, 0.f, 0.f);
}

extern "C" void kernel_launch(void* const* d_in, const int* in_sizes, int n_in,
                              void* d_out, int out_size, void* d_ws, size_t ws_size,
                              hipStream_t stream) {
    (void)in_sizes; (void)n_in; (void)out_size; (void)ws_size;

    const float* x            = (const float*)d_in[0];
    const float* D_v_beta     = (const float*)d_in[1];
    const float* D_e_beta_inv = (const float*)d_in[2];
    const float* D_e_alpha    = (const float*)d_in[3];
    const float* D_v_alpha_inv= (const float*)d_in[4];
    const float* W1_v2e = (const float*)d_in[5];
    const float* b1_v2e = (const float*)d_in[6];
    const float* W1_e2v = (const float*)d_in[7];
    const float* b1_e2v = (const float*)d_in[8];
    const float* W2_v2e = (const float*)d_in[9];
    const float* b2_v2e = (const float*)d_in[10];
    const float* W2_e2v = (const float*)d_in[11];
    const float* b2_e2v = (const float*)d_in[12];
    const int*   node_idx = (const int*)d_in[13];
    const int*   edge_idx = (const int*)d_in[14];
    float* outp = (float*)d_out;

    float* ws   = (float*)d_ws;
    float* H    = ws;                                  // N*D (H1, reused as H3)
    float* Eacc = H    + (size_t)N_NODES * DIM;        // M*D (accumulator)
    float* E2   = Eacc + (size_t)M_EDGES * DIM;        // M*D
    float* H2   = E2   + (size_t)M_EDGES * DIM;        // N*D (layer-1 output)

    const dim3 blk(256);
    const int gN = (N_NODES + ROWS_PER_BLOCK - 1) / ROWS_PER_BLOCK;   // 1563
    const int gM = (M_EDGES + ROWS_PER_BLOCK - 1) / ROWS_PER_BLOCK;   // 625
    const int gS = (E_INC + 7) / 8;                                   // 80000
    const int zN = (N_NODES * DIM / 4 + 255) / 256;
    const int zM = (M_EDGES * DIM / 4 + 255) / 256;

    // ===== Layer 1 =====
    // H1 = D_v_beta * (x @ W1_v2e + b1)
    hnhn_gemm_kernel<false><<<gN, blk, 0, stream>>>(x, W1_v2e, b1_v2e, D_v_beta, H, N_NODES);
    // Eacc = segment_sum(D_e_beta_inv[edge] * H1[node])
    hnhn_zero_kernel<<<zM, blk, 0, stream>>>(Eacc, M_EDGES * DIM);
    hnhn_scatter_kernel<<<gS, blk, 0, stream>>>(H, node_idx, edge_idx, D_e_beta_inv, Eacc, E_INC);
    // E2 = D_e_alpha * (relu(Eacc) @ W1_e2v + b1_e2v)
    hnhn_gemm_kernel<true><<<gM, blk, 0, stream>>>(Eacc, W1_e2v, b1_e2v, D_e_alpha, E2, M_EDGES);
    // H2 = segment_sum(D_v_alpha_inv[node] * E2[edge])
    hnhn_zero_kernel<<<zN, blk, 0, stream>>>(H2, N_NODES * DIM);
    hnhn_scatter_kernel<<<gS, blk, 0, stream>>>(E2, edge_idx, node_idx, D_v_alpha_inv, H2, E_INC);

    // ===== Layer 2 (inter-layer relu folded into GEMM A-load) =====
    // H3 = D_v_beta * (relu(H2) @ W2_v2e + b2)
    hnhn_gemm_kernel<true><<<gN, blk, 0, stream>>>(H2, W2_v2e, b2_v2e, D_v_beta, H, N_NODES);
    hnhn_zero_kernel<<<zM, blk, 0, stream>>>(Eacc, M_EDGES * DIM);
    hnhn_scatter_kernel<<<gS, blk, 0, stream>>>(H, node_idx, edge_idx, D_e_beta_inv, Eacc, E_INC);
    hnhn_gemm_kernel<true><<<gM, blk, 0, stream>>>(Eacc, W2_e2v, b2_e2v, D_e_alpha, E2, M_EDGES);
    hnhn_zero_kernel<<<zN, blk, 0, stream>>>(outp, N_NODES * DIM);
    hnhn_scatter_kernel<<<gS, blk, 0, stream>>>(E2, edge_idx, node_idx, D_v_alpha_inv, outp, E_INC);
}